// ConvCNP_463856468358
// MI455X (gfx1250) — compile-verified
//
#include <hip/hip_runtime.h>

#define LOG2E 1.4426950408889634f

typedef __attribute__((ext_vector_type(2))) float v2f;
typedef __attribute__((ext_vector_type(8))) float v8f;

// Problem constants
#define BB   8
#define NN   1024
#define MM   1024
#define CIN  8
#define CREP 9
#define COUT 16
#define MT   32          // m-tile per block
#define NCHUNK 8         // N split across 8 waves
#define NPER (NN / NCHUNK)

__global__ __launch_bounds__(256) void convcnp_kernel(
    const float* __restrict__ x,     // (B, N, 1)
    const float* __restrict__ y,     // (B, N, CIN)
    const float* __restrict__ t,     // (B, M, 1)
    const float* __restrict__ sigma, // (CREP,)
    const float* __restrict__ W,     // (COUT, CREP)
    const float* __restrict__ bias,  // (COUT,)
    float* __restrict__ out)         // (B, M, COUT)
{
    __shared__ float  sx[NN];             // x for this batch
    __shared__ float4 sy4[NN * 2];        // y for this batch (8 floats / n)
    __shared__ float  spart[NCHUNK * MT * 10]; // partial sums (pad 10)
    __shared__ float  yo[MT * 12];        // normalized y_out, K padded to 12
    __shared__ float  Wp[COUT * 12];      // W padded to K=12

    const int tid   = threadIdx.x;
    const int b     = blockIdx.y;
    const int mt    = blockIdx.x;
    const int m_l   = tid & (MT - 1);     // lane-owned m within tile
    const int chunk = tid >> 5;           // wave id == n-chunk id

    // ---- stage x, y into LDS (vectorized) ----
    {
        const float4* x4 = (const float4*)(x + b * NN);
        ((float4*)sx)[tid] = x4[tid];
        const float4* y4 = (const float4*)(y + (size_t)b * NN * CIN);
        #pragma unroll
        for (int k = 0; k < 8; ++k)
            sy4[tid + k * 256] = y4[tid + k * 256];
    }

    // per-channel exponent coefficient: arg = d * k2[c], exp2-based
    // k2[c] = -0.5*log2(e)*exp(-2*sigma_c)
    float k2[CREP];
    #pragma unroll
    for (int c = 0; c < CREP; ++c) {
        float s = sigma[c];
        k2[c] = -0.5f * LOG2E * __builtin_amdgcn_exp2f(-2.0f * s * LOG2E);
    }

    const float tm = t[b * MM + mt * MT + m_l];
    __syncthreads();

    // ---- main exp-weighted reduction over this wave's n-chunk ----
    float acc[CREP];
    #pragma unroll
    for (int c = 0; c < CREP; ++c) acc[c] = 0.f;

    const int n0 = chunk * NPER;
    #pragma unroll 4
    for (int i = 0; i < NPER; ++i) {
        const int n = n0 + i;
        const float  xn = sx[n];            // broadcast within wave
        const float4 ya = sy4[2 * n];       // broadcast
        const float4 yb = sy4[2 * n + 1];   // broadcast
        const float diff = xn - tm;
        const float dd   = diff * diff;
        float yr[CIN] = { ya.x, ya.y, ya.z, ya.w, yb.x, yb.y, yb.z, yb.w };
        acc[0] += __builtin_amdgcn_exp2f(dd * k2[0]);
        #pragma unroll
        for (int c = 1; c < CREP; ++c)
            acc[c] += __builtin_amdgcn_exp2f(dd * k2[c]) * yr[c - 1];
    }

    // ---- cross-chunk reduction via LDS ----
    {
        const int base = (chunk * MT + m_l) * 10;
        #pragma unroll
        for (int c = 0; c < CREP; ++c) spart[base + c] = acc[c];
    }
    __syncthreads();

    if (tid < MT) {                       // wave 0, fully active
        float s[CREP];
        #pragma unroll
        for (int c = 0; c < CREP; ++c) s[c] = 0.f;
        for (int ch = 0; ch < NCHUNK; ++ch) {
            const int base = (ch * MT + tid) * 10;
            #pragma unroll
            for (int c = 0; c < CREP; ++c) s[c] += spart[base + c];
        }
        const float dens = s[0];
        const float inv  = 1.0f / (dens + 1e-8f);
        yo[tid * 12 + 0] = dens;
        #pragma unroll
        for (int c = 1; c < CREP; ++c) yo[tid * 12 + c] = s[c] * inv;
        yo[tid * 12 + 9]  = 0.f;
        yo[tid * 12 + 10] = 0.f;
        yo[tid * 12 + 11] = 0.f;
    }
    if (tid < COUT * 12) {                // waves 0..5, fully active
        const int o = tid / 12, c = tid % 12;
        Wp[tid] = (c < CREP) ? W[o * CREP + c] : 0.f;
    }
    __syncthreads();

    // ---- WMMA epilogue: out[m(32) x o(16)] = yo(32x12) * Wp^T(12x16) ----
    const int wid  = tid >> 5;
    const int lane = tid & 31;
    if (wid < 2) {
        const int half  = lane >> 4;      // 0: K pair {0,1}; 1: K pair {2,3}
        const int lr    = lane & 15;
        const int mbase = wid * 16;

        v8f d = {0.f, 0.f, 0.f, 0.f, 0.f, 0.f, 0.f, 0.f};
        #pragma unroll
        for (int kc = 0; kc < 3; ++kc) {
            const int c0 = kc * 4 + 2 * half;
            v2f a, bm;
            a.x  = yo[(mbase + lr) * 12 + c0];
            a.y  = yo[(mbase + lr) * 12 + c0 + 1];
            bm.x = Wp[lr * 12 + c0];
            bm.y = Wp[lr * 12 + c0 + 1];
            d = __builtin_amdgcn_wmma_f32_16x16x4_f32(
                    false, a, false, bm, (short)0, d, false, false);
        }

        const float bo = bias[lr];
        const int mg = mt * MT + mbase;
        #pragma unroll
        for (int r = 0; r < 8; ++r) {
            const int m = mg + r + 8 * half;
            out[((size_t)b * MM + m) * COUT + lr] = d[r] + bo;
        }
    }
}

extern "C" void kernel_launch(void* const* d_in, const int* in_sizes, int n_in,
                              void* d_out, int out_size, void* d_ws, size_t ws_size,
                              hipStream_t stream) {
    const float* x     = (const float*)d_in[0];
    const float* y     = (const float*)d_in[1];
    const float* t     = (const float*)d_in[2];
    const float* sigma = (const float*)d_in[3];
    const float* W     = (const float*)d_in[4];
    const float* bias  = (const float*)d_in[5];
    float* out = (float*)d_out;

    dim3 grid(MM / MT, BB);   // 32 x 8 = 256 blocks
    dim3 block(256);
    convcnp_kernel<<<grid, block, 0, stream>>>(x, y, t, sigma, W, bias, out);
}